// CausalStateSpaceModel_71906342469948
// MI455X (gfx1250) — compile-verified
//
#include <hip/hip_runtime.h>
#include <hip/hip_bf16.h>

// ---------------------------------------------------------------------------
// Causal SSM on MI455X (gfx1250), fp32 WMMA (V_WMMA_F32_16X16X4_F32) + TDM.
//
// Roofline: ~15 GFLOP fp32 vs ~200 MB HBM traffic -> memory bound (~9us at
// 23.3 TB/s). fp32 matrix cores because the 8192-step recurrence with
// A ~ I + 0.01N cannot tolerate bf16/f16 state accumulation.
//
// Chunked parallel scan (L = 32, NC = 256 chunks):
//   state_{cL+j} = S_{c-1} (A^T)^{j+1} + local_j,  local = zero-init chunk scan
//   out = local @ C^T + x @ D  (written by k_scan)
//       + S_{c-1} @ R_j,  R_j = (A^T)^{j+1} C^T  (added by k_correct)
//
// k_scan stages each x[:,t,:] tile with TENSOR_LOAD_TO_LDS (double-buffered,
// DMA for step j+1 overlaps the WMMA GEMMs of step j, completion via
// s_wait_tensorcnt on the issuing wave).
// ---------------------------------------------------------------------------

typedef __attribute__((ext_vector_type(2))) float v2f;
typedef __attribute__((ext_vector_type(8))) float v8f;
typedef __attribute__((ext_vector_type(4))) unsigned int u32x4;
typedef __attribute__((ext_vector_type(4))) int i32x4;
typedef __attribute__((ext_vector_type(8))) int i32x8;

static constexpr int L  = 32;        // chunk length
static constexpr int NC = 8192 / L;  // 256 chunks
static constexpr int BATCH = 32;
static constexpr int DIM = 64;

__device__ __forceinline__ v8f wmma4(v2f a, v2f b, v8f c) {
  // V_WMMA_F32_16X16X4_F32 : D = A(16x4) * B(4x16) + C(16x16)
  return __builtin_amdgcn_wmma_f32_16x16x4_f32(
      false, a, false, b, (short)0, c, false, false);
}

// A-operand (16x4 f32): lane m = lane&15, VGPR pair holds k = kbase + 2*(lane>>4) + {0,1}
// Source is row-major [m][64]; identical index math serves the B-operand when
// the weight is staged in [n][k] layout. 8-byte aligned float2 loads.
__device__ __forceinline__ v2f ld_op(const float* base, int lane, int kbase) {
  const float* p = base + ((lane & 15) << 6) + kbase + ((lane >> 4) << 1);
  return *(const v2f*)p;
}

// C/D tile (16x16 f32, 8 VGPRs): element (r + 8*(lane>=16), lane&15).
__device__ __forceinline__ void store_tile(float* dst, int lane, v8f c) {
  int n = lane & 15;
  int mh = (lane >> 4) << 3;
#pragma unroll
  for (int r = 0; r < 8; ++r) dst[(r + mh) * DIM + n] = c[r];
}

// ---------------------------------------------------------------------------
// Tensor Data Mover: async 2D tile load global -> LDS.
// Descriptor group0: {count=1 (valid, load, user), lds_addr, gaddr_lo,
//                     gaddr_hi[24:0] | type=2<<30}.
// Descriptor group1: workgroup_mask=0, data_size=2 (4B),
//                    tensor_dim0=64 | tile_dim0=64, tensor_dim1=32 | tile_dim1=32,
//                    tensor_dim0_stride = 8192*64 elements (batch stride).
// Groups 2/3 zero (tile_dim2/3 = 0 = unused -> 2D).
// ---------------------------------------------------------------------------
__device__ __forceinline__ void tdm_load_2d(u32x4 g0, i32x8 g1) {
#if __has_include(<hip/amd_detail/amd_gfx1250_TDM.h>)
  // amdgpu-toolchain (clang-23 / therock-10.0): 6-arg form
  __builtin_amdgcn_tensor_load_to_lds(g0, g1, (i32x4)0, (i32x4)0, (i32x8)0, 0);
#else
  // ROCm 7.2 (clang-22): 5-arg form
  __builtin_amdgcn_tensor_load_to_lds(g0, g1, (i32x4)0, (i32x4)0, 0);
#endif
}

__device__ __forceinline__ void tdm_issue_x(const float* gptr, unsigned lds_off,
                                            const i32x8& g1) {
  unsigned long long ga = (unsigned long long)gptr;
  u32x4 g0;
  g0.x = 1u;                                              // count=1, load, user
  g0.y = lds_off;                                         // LDS byte address
  g0.z = (unsigned)ga;                                    // global addr [31:0]
  g0.w = ((unsigned)(ga >> 32) & 0x01FFFFFFu) | 0x80000000u;  // [56:32] | type=2
  tdm_load_2d(g0, g1);
}

// ---------------------------------------------------------------------------
// K1: P_m = (A^T)^m, m = 1..L, stored at P + (m-1)*4096, row-major [a][b].
// ---------------------------------------------------------------------------
__global__ void k_powers(const float* __restrict__ Aw, float* __restrict__ P) {
  __shared__ float sA[DIM * DIM];    // A row-major == [n][k] layout of A^T (B-operand)
  __shared__ float sBuf0[DIM * DIM]; // current power, row-major (A-operand)
  __shared__ float sBuf1[DIM * DIM];
  int tid = threadIdx.x, lane = tid & 31, wave = tid >> 5;

  for (int i = tid; i < DIM * DIM; i += 256) {
    sA[i] = Aw[i];
    int m = i >> 6, k = i & 63;
    float at = Aw[k * DIM + m];      // A^T[m][k]
    sBuf0[i] = at;                   // P_1
    P[i] = at;
  }
  __syncthreads();

  float* cur = sBuf0;
  float* nxt = sBuf1;
  for (int m = 2; m <= L; ++m) {
#pragma unroll
    for (int rep = 0; rep < 2; ++rep) {
      int tt = wave + rep * 8;       // 16 tiles / 8 waves
      int mt = tt >> 2, nt = tt & 3;
      v8f acc = {};
      const float* arow = cur + mt * 16 * DIM;
      const float* bcol = sA + nt * 16 * DIM;
#pragma unroll
      for (int kt = 0; kt < 16; ++kt)
        acc = wmma4(ld_op(arow, lane, kt * 4), ld_op(bcol, lane, kt * 4), acc);
      store_tile(nxt + mt * 16 * DIM + nt * 16, lane, acc);
      store_tile(P + (m - 1) * 4096 + mt * 16 * DIM + nt * 16, lane, acc);
    }
    __syncthreads();
    float* t = cur; cur = nxt; nxt = t;
  }
}

// ---------------------------------------------------------------------------
// K2: R_j = P_{j+1} @ C^T, stored TRANSPOSED as Rt[j][n][a] ([n][k] B-operand layout)
// ---------------------------------------------------------------------------
__global__ void k_rmats(const float* __restrict__ Cw, const float* __restrict__ P,
                        float* __restrict__ Rt) {
  int j = blockIdx.x;  // 0..L-1 ; P_{j+1} at offset j*4096
  __shared__ float sP[DIM * DIM];
  __shared__ float sC[DIM * DIM];  // C row-major == [n][k] layout of C^T
  int tid = threadIdx.x, lane = tid & 31, wave = tid >> 5;
  for (int i = tid; i < DIM * DIM; i += 256) {
    sP[i] = P[j * 4096 + i];
    sC[i] = Cw[i];
  }
  __syncthreads();
#pragma unroll
  for (int rep = 0; rep < 2; ++rep) {
    int tt = wave + rep * 8;
    int mt = tt >> 2, nt = tt & 3;
    v8f acc = {};
#pragma unroll
    for (int kt = 0; kt < 16; ++kt)
      acc = wmma4(ld_op(sP + mt * 16 * DIM, lane, kt * 4),
                  ld_op(sC + nt * 16 * DIM, lane, kt * 4), acc);
    int n = nt * 16 + (lane & 15);
    int mh = (lane >> 4) << 3;
#pragma unroll
    for (int r = 0; r < 8; ++r) {
      int a = mt * 16 + r + mh;
      Rt[(long)j * 4096 + n * DIM + a] = acc[r];
    }
  }
}

// ---------------------------------------------------------------------------
// K3: per-chunk local scan (zero init) over all 32 batches, fused with
//     Bu = x @ B^T  and partial output  out = local @ C^T + x @ D.
//     One workgroup per chunk; 8 waves own the 8 16x16 tiles of the
//     [32 x 64] state. x tiles staged by TDM, double-buffered.
// ---------------------------------------------------------------------------
__global__ void k_scan(const float* __restrict__ x, const float* __restrict__ Bw,
                       const float* __restrict__ Aw, const float* __restrict__ Cw,
                       const float* __restrict__ Dw, float* __restrict__ out,
                       float* __restrict__ F) {
  __shared__ float sB[DIM * DIM];   // B row-major == [n][i] layout of B^T
  __shared__ float sA[DIM * DIM];   // A row-major == [n][k] layout of A^T
  __shared__ float sC[DIM * DIM];   // C row-major == [n][k] layout of C^T
  __shared__ float sDt[DIM * DIM];  // D transposed -> [n][i] layout of D
  __shared__ float sX[2][BATCH * DIM]; // double-buffered x[:, t, :]
  __shared__ float sS[2][BATCH * DIM];

  int tid = threadIdx.x, lane = tid & 31, wave = tid >> 5;
  int mt = wave >> 2;   // batch-row tile 0..1
  int nt = wave & 3;    // column tile 0..3
  int c = blockIdx.x;   // chunk

  // TDM group1 (uniform constant): 4B elems, 64x32 tile == tensor window,
  // batch stride 8192*64 elements.
  const i32x8 g1 = { 0x00020000, 64 << 16, 32 << 16, 64 << 16,
                     32, 8192 * 64, 0, 0 };

  for (int i = tid; i < DIM * DIM; i += 256) {
    sB[i] = Bw[i];
    sA[i] = Aw[i];
    sC[i] = Cw[i];
    int k = i >> 6, n = i & 63;
    sDt[n * DIM + k] = Dw[i];
  }
  for (int i = tid; i < BATCH * DIM; i += 256) sS[0][i] = 0.f;

  // prologue: kick off DMA for step 0 into buffer 0
  if (wave == 0)
    tdm_issue_x(x + (long)c * L * DIM, (unsigned)(unsigned long long)&sX[0][0], g1);
  __syncthreads();

  const float* sBn = sB + nt * 16 * DIM;
  const float* sAn = sA + nt * 16 * DIM;
  const float* sCn = sC + nt * 16 * DIM;
  const float* sDn = sDt + nt * 16 * DIM;

  int cur = 0;
  for (int j = 0; j < L; ++j) {
    long t = (long)c * L + j;
    if (wave == 0)
      __builtin_amdgcn_s_wait_tensorcnt(0);  // x[j] tile landed in LDS
    __syncthreads();                         // release all waves
    if (wave == 0 && j + 1 < L)              // overlap DMA(j+1) with compute(j)
      tdm_issue_x(x + (t + 1) * DIM,
                  (unsigned)(unsigned long long)&sX[(j + 1) & 1][0], g1);

    const float* xrow = sX[j & 1] + mt * 16 * DIM;
    // Bu tile = x @ B^T
    v8f acc = {};
#pragma unroll
    for (int kt = 0; kt < 16; ++kt)
      acc = wmma4(ld_op(xrow, lane, kt * 4), ld_op(sBn, lane, kt * 4), acc);
    // state = state @ A^T + Bu
    const float* srow = sS[cur] + mt * 16 * DIM;
#pragma unroll
    for (int kt = 0; kt < 16; ++kt)
      acc = wmma4(ld_op(srow, lane, kt * 4), ld_op(sAn, lane, kt * 4), acc);

    int nxt = cur ^ 1;
    store_tile(sS[nxt] + mt * 16 * DIM + nt * 16, lane, acc);
    __syncthreads();

    // partial out = newstate @ C^T + x @ D
    v8f o = {};
    const float* nsrow = sS[nxt] + mt * 16 * DIM;
#pragma unroll
    for (int kt = 0; kt < 16; ++kt) {
      o = wmma4(ld_op(nsrow, lane, kt * 4), ld_op(sCn, lane, kt * 4), o);
      o = wmma4(ld_op(xrow, lane, kt * 4), ld_op(sDn, lane, kt * 4), o);
    }
    {
      int n = nt * 16 + (lane & 15);
      int mh = (lane >> 4) << 3;
#pragma unroll
      for (int r = 0; r < 8; ++r) {
        int b = mt * 16 + r + mh;
        out[((long)b * 8192 + t) * DIM + n] = o[r];
      }
    }
    cur = nxt;
    __syncthreads();
  }
  // final chunk state -> F[c][b][n]
  {
    int n = nt * 16 + (lane & 15);
    int mh = (lane >> 4) << 3;
    const float* s = sS[cur];
#pragma unroll
    for (int r = 0; r < 8; ++r) {
      int b = mt * 16 + r + mh;
      F[((long)c * BATCH + b) * DIM + n] = s[b * DIM + n];
    }
  }
}

// ---------------------------------------------------------------------------
// K4: serial carry scan over chunks.  S_c = S_{c-1} @ Q + F_c, Q = (A^T)^L.
//     CARRY[c] = S_{c-1} (carry-in for chunk c).
// ---------------------------------------------------------------------------
__global__ void k_carry(const float* __restrict__ P, const float* __restrict__ F,
                        float* __restrict__ CARRY) {
  __shared__ float sQ[DIM * DIM];        // [n][k] layout of Q
  __shared__ float sS[2][BATCH * DIM];
  int tid = threadIdx.x, lane = tid & 31, wave = tid >> 5;
  int mt = wave >> 2, nt = wave & 3;
  for (int i = tid; i < DIM * DIM; i += 256) {
    int k = i >> 6, n = i & 63;
    sQ[n * DIM + k] = P[(L - 1) * 4096 + i];  // transpose Q into B-operand layout
  }
  for (int i = tid; i < BATCH * DIM; i += 256) sS[0][i] = 0.f;
  __syncthreads();

  int cur = 0;
  for (int c = 0; c < NC; ++c) {
    int n = nt * 16 + (lane & 15);
    int mh = (lane >> 4) << 3;
    const float* s = sS[cur];
#pragma unroll
    for (int r = 0; r < 8; ++r) {   // carry-in for chunk c = current state
      int b = mt * 16 + r + mh;
      CARRY[((long)c * BATCH + b) * DIM + n] = s[b * DIM + n];
    }
    v8f acc{};
#pragma unroll
    for (int r = 0; r < 8; ++r) {   // accumulator init = F_c tile
      int b = mt * 16 + r + mh;
      acc[r] = F[((long)c * BATCH + b) * DIM + n];
    }
#pragma unroll
    for (int kt = 0; kt < 16; ++kt)
      acc = wmma4(ld_op(sS[cur] + mt * 16 * DIM, lane, kt * 4),
                  ld_op(sQ + nt * 16 * DIM, lane, kt * 4), acc);
    int nxt = cur ^ 1;
    store_tile(sS[nxt] + mt * 16 * DIM + nt * 16, lane, acc);
    __syncthreads();
    cur = nxt;
  }
}

// ---------------------------------------------------------------------------
// K5: out[b, c*L + j, :] += CARRY[c][b] @ R_j   (fully parallel WMMA)
// ---------------------------------------------------------------------------
__global__ void k_correct(const float* __restrict__ CARRY, const float* __restrict__ Rt,
                          float* __restrict__ out) {
  int c = blockIdx.x;     // chunk
  int jq = blockIdx.y;    // j-quarter
  __shared__ float sCar[BATCH * DIM];  // row-major [b][k] (A-operand)
  int tid = threadIdx.x, lane = tid & 31, wave = tid >> 5;
  int mt = wave >> 2, nt = wave & 3;
  for (int i = tid; i < BATCH * DIM; i += 256)
    sCar[i] = CARRY[(long)c * BATCH * DIM + i];
  __syncthreads();

  for (int jj = 0; jj < L / 4; ++jj) {
    int j = jq * (L / 4) + jj;
    long t = (long)c * L + j;
    const float* rb = Rt + (long)j * 4096 + nt * 16 * DIM;  // global [n][k] B-operand
    v8f acc = {};
#pragma unroll
    for (int kt = 0; kt < 16; ++kt)
      acc = wmma4(ld_op(sCar + mt * 16 * DIM, lane, kt * 4),
                  ld_op(rb, lane, kt * 4), acc);
    int n = nt * 16 + (lane & 15);
    int mh = (lane >> 4) << 3;
#pragma unroll
    for (int r = 0; r < 8; ++r) {
      int b = mt * 16 + r + mh;
      long idx = ((long)b * 8192 + t) * DIM + n;
      out[idx] += acc[r];
    }
  }
}

// ---------------------------------------------------------------------------
extern "C" void kernel_launch(void* const* d_in, const int* in_sizes, int n_in,
                              void* d_out, int out_size, void* d_ws, size_t ws_size,
                              hipStream_t stream) {
  (void)in_sizes; (void)n_in; (void)out_size; (void)ws_size;
  const float* x  = (const float*)d_in[0];
  const float* Aw = (const float*)d_in[1];
  const float* Bw = (const float*)d_in[2];
  const float* Cw = (const float*)d_in[3];
  const float* Dw = (const float*)d_in[4];
  float* out = (float*)d_out;
  float* ws  = (float*)d_ws;

  float* P   = ws;                          // L * 4096 floats
  float* Rt  = P + (long)L * 4096;          // L * 4096
  float* F   = Rt + (long)L * 4096;         // NC * 32 * 64
  float* CAR = F + (long)NC * BATCH * DIM;  // NC * 32 * 64
  // total ws: (2*L*4096 + 2*NC*2048) * 4B ~= 5.2 MB

  k_powers<<<1, 256, 0, stream>>>(Aw, P);
  k_rmats<<<L, 256, 0, stream>>>(Cw, P, Rt);
  k_scan<<<NC, 256, 0, stream>>>(x, Bw, Aw, Cw, Dw, out, F);
  k_carry<<<1, 256, 0, stream>>>(P, F, CAR);
  k_correct<<<dim3(NC, 4), 256, 0, stream>>>(CAR, Rt, out);
}